// GraphConvLayer_4629974745886
// MI455X (gfx1250) — compile-verified
//
#include <hip/hip_runtime.h>

// ---------------- problem constants ----------------
#define NN      10000
#define EE      320000
#define FDIM    128
#define WUSED   313      // ceil(10000/32)
#define WPR     320      // padded words per bitmap row
#define BN_EPS  1e-5f
#define DEG_EPS 1e-8f

typedef __attribute__((ext_vector_type(2))) float v2f;
typedef __attribute__((ext_vector_type(8))) float v8f;

// ---------------- 1: zero bitmap + BN accumulators ----------------
__global__ __launch_bounds__(256) void k_zero(unsigned* __restrict__ bitmap,
                                              float* __restrict__ stats) {
    size_t i = (size_t)blockIdx.x * 256 + threadIdx.x;
    if (i < (size_t)NN * WPR) bitmap[i] = 0u;
    if (blockIdx.x == 0) stats[threadIdx.x] = 0.0f;   // 256 floats: sum, sumsq
}

// ---------------- 2: scatter edges into bit matrix (dedup = set semantics) ----
__global__ __launch_bounds__(256) void k_edges(const int* __restrict__ ei,
                                               unsigned* __restrict__ bitmap) {
    int e = blockIdx.x * 256 + threadIdx.x;
    if (e >= EE) return;
    int r = ei[e];          // edge_index[0][e]
    int c = ei[EE + e];     // edge_index[1][e]
    atomicOr(&bitmap[(size_t)r * WPR + (c >> 5)], 1u << (c & 31));
}

// ---------------- 3: degree -> d^{-1/2} (one wave32 per row) ----------------
__global__ __launch_bounds__(128) void k_degree(const unsigned* __restrict__ bitmap,
                                                float* __restrict__ dinv) {
    int wave = threadIdx.x >> 5, lane = threadIdx.x & 31;
    int r = blockIdx.x * 4 + wave;
    if (r >= NN) return;
    const unsigned* row = bitmap + (size_t)r * WPR;
    int cnt = 0;
    for (int w = lane; w < WUSED; w += 32) cnt += __popc(row[w]);
    for (int off = 16; off; off >>= 1) cnt += __shfl_xor(cnt, off, 32);
    // deg = edge-bits + 1 (the added identity); diag may be 2 total (bit + eye)
    if (lane == 0) dinv[r] = rsqrtf((float)(cnt + 1) + DEG_EPS);
}

// ---------------- 4: SpMM h1 = D^-1/2 (A+I) D^-1/2 x  (block per row) -------
__global__ __launch_bounds__(128) void k_spmm(const unsigned* __restrict__ bitmap,
                                              const float* __restrict__ dinv,
                                              const float* __restrict__ x,
                                              float* __restrict__ h1) {
    int r = blockIdx.x;
    int f = threadIdx.x;
    float dr = dinv[r];
    // identity term: contributes dinv[r]^2 * x[r]; factor one dinv[r] out
    float acc = dr * x[(size_t)r * FDIM + f];
    const unsigned* row = bitmap + (size_t)r * WPR;
    for (int w = 0; w < WUSED; ++w) {
        unsigned bits = row[w];               // uniform across block -> s_load
        while (bits) {
            int b = __ffs(bits) - 1;
            bits &= bits - 1;
            int c = w * 32 + b;
            acc += dinv[c] * x[(size_t)c * FDIM + f];
        }
    }
    h1[(size_t)r * FDIM + f] = dr * acc;
}

// ---------------- 5: WMMA f32 GEMM  h2 = h1 @ W + b -------------------------
// Block: 128 threads (4 waves). Block owns a 16-row stripe; wave w owns column
// tiles [32w, 32w+16) and [32w+16, 32w+32). K consumed in 32-wide slices that
// are staged with GLOBAL_LOAD_ASYNC_TO_LDS_B128 (ASYNCcnt path, no VGPR round
// trip). W slice stored row-major, pitch 144 floats (16B-aligned B128 stores,
// disjoint banks for the two K-rows of each B fragment). A pitch 36 floats:
// 36*m mod 64 distinct for m=0..15 -> conflict-free ds_load_b64 A fragments.
#define KC      32
#define APITCH  36    // floats (144 B rows)
#define WPITCH  144   // floats (576 B rows)

__global__ __launch_bounds__(128) void k_gemm(const float* __restrict__ h1,
                                              const float* __restrict__ W,
                                              const float* __restrict__ bias,
                                              float* __restrict__ h2) {
    __shared__ __align__(16) float Asl[16 * APITCH];   //  2304 B
    __shared__ __align__(16) float Wl[KC * WPITCH];    // 18432 B
    const int t    = threadIdx.x;
    const int wave = t >> 5, lane = t & 31;
    const int m0   = blockIdx.x * 16;
    const int mloc = lane & 15;          // A row / B col within 16-wide tile
    const int kb   = (lane >> 4) * 2;    // K-pair selector (16x4 f32 layout)
    const int nc0  = wave * 32;

    // per-lane async-staging geometry
    const int am   = (wave << 2) + (lane >> 3);   // A row this lane copies
    const int aseg = lane & 7;                    // 16B segment within A row
    const unsigned a_dst =
        (unsigned)(size_t)(void*)&Asl[am * APITCH + aseg * 4];
    const unsigned w_dst0 =
        (unsigned)(size_t)(void*)&Wl[(wave << 3) * WPITCH + lane * 4];

    v8f acc0 = {};
    v8f acc1 = {};

    for (int kc = 0; kc < FDIM; kc += KC) {
        // ---- async fill: A stripe (16x32) + W slice (32x128), LDS-direct ----
        {
            const float* ag = &h1[(size_t)(m0 + am) * FDIM + kc + aseg * 4];
            asm volatile("global_load_async_to_lds_b128 %0, %1, off"
                         :: "v"(a_dst), "v"(ag) : "memory");
        }
        #pragma unroll
        for (int i = 0; i < 8; ++i) {              // wave w copies W rows 8w..8w+7
            const float* wg = &W[(size_t)(kc + (wave << 3) + i) * FDIM + lane * 4];
            unsigned wd = w_dst0 + (unsigned)(i * WPITCH * 4);
            asm volatile("global_load_async_to_lds_b128 %0, %1, off"
                         :: "v"(wd), "v"(wg) : "memory");
        }
        asm volatile("s_wait_asynccnt 0x0" ::: "memory");
        __syncthreads();

        // ---- 8 WMMA K-steps per slice, two 16-col tiles per wave ----
        #pragma unroll
        for (int kk = 0; kk < KC; kk += 4) {
            v2f a = *(const v2f*)&Asl[mloc * APITCH + kk + kb];
            int rk = (kk + kb) * WPITCH;
            v2f b0, b1;
            b0.x = Wl[rk + nc0 + mloc];
            b0.y = Wl[rk + WPITCH + nc0 + mloc];
            b1.x = Wl[rk + nc0 + 16 + mloc];
            b1.y = Wl[rk + WPITCH + nc0 + 16 + mloc];
            acc0 = __builtin_amdgcn_wmma_f32_16x16x4_f32(false, a, false, b0,
                                                         (short)0, acc0, false, false);
            acc1 = __builtin_amdgcn_wmma_f32_16x16x4_f32(false, a, false, b1,
                                                         (short)0, acc1, false, false);
        }
        __syncthreads();
    }

    // epilogue: C/D 16x16 f32 layout -> lane(l&15)=col, vgpr v -> row v+8*(l>>4)
    int half = lane >> 4;
    int n    = lane & 15;
    int c0 = nc0 + n, c1 = nc0 + 16 + n;
    float bb0 = bias[c0], bb1 = bias[c1];
    #pragma unroll
    for (int v = 0; v < 8; ++v) {
        int m = m0 + v + 8 * half;
        h2[(size_t)m * FDIM + c0] = acc0[v] + bb0;
        h2[(size_t)m * FDIM + c1] = acc1[v] + bb1;
    }
}

// ---------------- 6: BN batch stats (sum, sumsq per feature) ----------------
__global__ __launch_bounds__(128) void k_stats(const float* __restrict__ h2,
                                               float* __restrict__ stats) {
    int f = threadIdx.x;
    float s = 0.0f, q = 0.0f;
    for (int r = blockIdx.x; r < NN; r += gridDim.x) {
        float v = h2[(size_t)r * FDIM + f];
        s += v; q += v * v;
    }
    atomicAdd(&stats[f], s);
    atomicAdd(&stats[FDIM + f], q);
}

// ---------------- 7: fold BN into scale/shift -------------------------------
__global__ __launch_bounds__(128) void k_finalize(float* __restrict__ stats,
                                                  const float* __restrict__ gamma,
                                                  const float* __restrict__ beta) {
    int f = threadIdx.x;
    const float invN = 1.0f / (float)NN;
    float mean = stats[f] * invN;
    float var  = stats[FDIM + f] * invN - mean * mean;   // biased, matches jnp.var
    float scl  = gamma[f] * rsqrtf(var + BN_EPS);
    stats[256 + f] = scl;
    stats[384 + f] = beta[f] - mean * scl;
}

// ---------------- 8: apply ---------------------------------------------------
__global__ __launch_bounds__(256) void k_apply(const float* __restrict__ h2,
                                               const float* __restrict__ stats,
                                               float* __restrict__ out) {
    size_t i = (size_t)blockIdx.x * 256 + threadIdx.x;
    if (i >= (size_t)NN * FDIM) return;
    int f = (int)(i & (FDIM - 1));
    out[i] = h2[i] * stats[256 + f] + stats[384 + f];
}

// ---------------- launch -----------------------------------------------------
extern "C" void kernel_launch(void* const* d_in, const int* in_sizes, int n_in,
                              void* d_out, int out_size, void* d_ws, size_t ws_size,
                              hipStream_t stream) {
    const float* x     = (const float*)d_in[0];
    const int*   ei    = (const int*)  d_in[1];
    const float* W     = (const float*)d_in[2];
    const float* bias  = (const float*)d_in[3];
    const float* gamma = (const float*)d_in[4];
    const float* beta  = (const float*)d_in[5];
    float* out = (float*)d_out;

    // workspace layout
    unsigned* bitmap = (unsigned*)d_ws;                               // NN*WPR words
    float* dinv  = (float*)((char*)d_ws + (size_t)NN * WPR * 4);      // 10016 floats (padded)
    float* h1    = dinv + 10016;                                      // NN*FDIM
    float* h2    = h1 + (size_t)NN * FDIM;                            // NN*FDIM
    float* stats = h2 + (size_t)NN * FDIM;                            // 512 floats

    size_t nwords = (size_t)NN * WPR;
    k_zero   <<<(int)((nwords + 255) / 256), 256, 0, stream>>>(bitmap, stats);
    k_edges  <<<(EE + 255) / 256,            256, 0, stream>>>(ei, bitmap);
    k_degree <<<(NN + 3) / 4,                128, 0, stream>>>(bitmap, dinv);
    k_spmm   <<<NN,                          128, 0, stream>>>(bitmap, dinv, x, h1);
    k_gemm   <<<NN / 16,                     128, 0, stream>>>(h1, W, bias, h2);
    k_stats  <<<100,                         128, 0, stream>>>(h2, stats);
    k_finalize<<<1,                          128, 0, stream>>>(stats, gamma, beta);
    k_apply  <<<(NN * FDIM + 255) / 256,     256, 0, stream>>>(h2, stats, out);
}